// ContourDistanceNormalLoss_51788715655266
// MI455X (gfx1250) — compile-verified
//
#include <hip/hip_runtime.h>
#include <cmath>

typedef __attribute__((ext_vector_type(2))) float v2f;
typedef __attribute__((ext_vector_type(8))) float v8f;

#define NSAMP 100
#define BLOCK 256

// Bilinear sample with border clamp; x,y are already clamped to [0, W-1]/[0, H-1]
// (clamps retained on the +1 taps per the reference's jnp.clip on x0+1 / y0+1).
__device__ __forceinline__ float bilin(const float* __restrict__ img, float x, float y,
                                       int W, float wmax, float hmax)
{
    float x0f = floorf(x);
    float y0f = floorf(y);
    float wx = x - x0f;
    float wy = y - y0f;
    int x0 = (int)fminf(fmaxf(x0f,       0.f), wmax);
    int x1 = (int)fminf(fmaxf(x0f + 1.f, 0.f), wmax);
    int y0 = (int)fminf(fmaxf(y0f,       0.f), hmax);
    int y1 = (int)fminf(fmaxf(y0f + 1.f, 0.f), hmax);
    float v00 = img[y0 * W + x0];
    float v01 = img[y0 * W + x1];
    float v10 = img[y1 * W + x0];
    float v11 = img[y1 * W + x1];
    float ax = 1.f - wx, ay = 1.f - wy;
    return v00 * ax * ay + v01 * wx * ay + v10 * ax * wy + v11 * wx * wy;
}

__global__ void __launch_bounds__(BLOCK)
contour_main(const float* __restrict__ img,
             const float* __restrict__ points,
             const float* __restrict__ normals,
             float* __restrict__ partials,
             int N, int W, int H, int nBlocks, int paddedBlocks)
{
    const int tid = threadIdx.x;
    const int gid = blockIdx.x * BLOCK + tid;
    const float wmax = (float)(W - 1);
    const float hmax = (float)(H - 1);

    float sq = 0.f;
    if (gid < N) {
        float2 p  = ((const float2*)points)[gid];
        float2 nm = ((const float2*)normals)[gid];

        // pts = clip(points, 0, max(H-1, W-1))  (H==W here)
        float px = fminf(fmaxf(p.x, 0.f), wmax);
        float py = fminf(fmaxf(p.y, 0.f), wmax);

        // d = normalize([-ny, nx])
        float dx = -nm.y, dy = nm.x;
        float nrm = sqrtf(dx * dx + dy * dy);
        dx /= nrm;
        dy /= nrm;

        const float MAXF = 3.4028234663852886e38f; // finfo(float32).max
        bool zx = (dx == 0.f), zy = (dy == 0.f);
        float sdx = zx ? 1.f : dx;
        float sdy = zy ? 1.f : dy;
        float t_left   = zx ? -MAXF : (0.f  - px) / sdx;
        float t_right  = zx ?  MAXF : (wmax - px) / sdx;
        float t_top    = zy ? -MAXF : (0.f  - py) / sdy;
        float t_bottom = zy ?  MAXF : (hmax - py) / sdy;
        float tmin = fmaxf(t_left, t_top);
        float tmax = fminf(t_right, t_bottom);

        float p1x = px + tmin * dx, p1y = py + tmin * dy;
        float p2x = px + tmax * dx, p2y = py + tmax * dy;
        float ex = p2x - p1x, ey = p2y - p1y;
        const float step = 1.f / (float)(NSAMP - 1);

        // streaming 3-tap local-minimum scan: vals[i-1] < vals[i-2] && vals[i-1] < vals[i]
        // i = 0
        float lx0 = fminf(fmaxf(p1x, 0.f), wmax);
        float ly0 = fminf(fmaxf(p1y, 0.f), hmax);
        float vprev = bilin(img, lx0, ly0, W, wmax, hmax);
        // i = 1
        float t  = step;
        float cx = fminf(fmaxf(p1x + t * ex, 0.f), wmax);
        float cy = fminf(fmaxf(p1y + t * ey, 0.f), hmax);
        float vcur = bilin(img, cx, cy, W, wmax, hmax);

        float best_val  = vcur;               // argmin over all-inf -> index 0 -> vals[1]
        float best_dist = __builtin_inff();

        #pragma unroll 4
        for (int i = 2; i < NSAMP; ++i) {
            t = (float)i * step;
            float nx_ = fminf(fmaxf(p1x + t * ex, 0.f), wmax);
            float ny_ = fminf(fmaxf(p1y + t * ey, 0.f), hmax);
            float vnext = bilin(img, nx_, ny_, W, wmax, hmax);
            if (vcur < vprev && vcur < vnext) {
                float ddx = cx - px, ddy = cy - py;   // dist uses the CLAMPED line coords
                float dist = sqrtf(ddx * ddx + ddy * ddy);
                if (dist < best_dist) { best_dist = dist; best_val = vcur; } // strict < keeps first (argmin)
            }
            vprev = vcur; vcur = vnext; cx = nx_; cy = ny_;
        }

        float ref_val = bilin(img, px, py, W, wmax, hmax);
        float e = best_val - ref_val;
        sq = e * e;
    }

    // zero the pad tail of the partials array so the WMMA reduce wave runs divergence-free
    if (blockIdx.x == 0 && tid < (paddedBlocks - nBlocks))
        partials[nBlocks + tid] = 0.f;

    // deterministic block tree-reduction in LDS
    __shared__ float sm[BLOCK];
    sm[tid] = sq;
    __syncthreads();
    for (int s = BLOCK / 2; s > 0; s >>= 1) {
        if (tid < s) sm[tid] += sm[tid + s];
        __syncthreads();
    }
    if (tid == 0) partials[blockIdx.x] = sm[0];
}

// Single-wave fp32 reduction using V_WMMA_F32_16X16X4_F32 with B == ones:
// D[m][n] = sum_k A[m][k] + C[m][n]  -> per-row running sums accumulated in C.
// A 16x4 fp32 layout: lanes 0-15 hold M=lane, K=0,1 in v[0],v[1]; lanes 16-31 K=2,3.
// D layout: lanes 0-15 VGPR r = row r (N=lane); lanes 16-31 VGPR r = row r+8.
__global__ void __launch_bounds__(32)
contour_reduce(const float* __restrict__ partials, int paddedN, float invN,
               float* __restrict__ out)
{
    const int lane = threadIdx.x;
    v8f c = {0.f, 0.f, 0.f, 0.f, 0.f, 0.f, 0.f, 0.f};
    v2f bones;
    bones.x = 1.f;
    bones.y = 1.f;
    const int m  = lane & 15;
    const int kb = (lane >> 4) << 1;           // K base: 0 for lanes 0-15, 2 for 16-31

    for (int base = 0; base < paddedN; base += 64) {
        const float* src = partials + base + m * 4 + kb;  // 8B-aligned
        v2f a = *(const v2f*)src;
        c = __builtin_amdgcn_wmma_f32_16x16x4_f32(
                /*neg_a=*/false, a, /*neg_b=*/false, bones,
                /*c_mod=*/(short)0, c, /*reuse_a=*/false, /*reuse_b=*/false);
    }

    // every lane's column n holds identical row sums; fold the 8 rows this lane sees
    float s = c[0] + c[1] + c[2] + c[3] + c[4] + c[5] + c[6] + c[7];
    // rows 0-7 live in lanes 0-15, rows 8-15 in lanes 16-31
    float total = __shfl(s, 0, 32) + __shfl(s, 16, 32);
    if (lane == 0) out[0] = total * invN;
}

extern "C" void kernel_launch(void* const* d_in, const int* in_sizes, int n_in,
                              void* d_out, int out_size, void* d_ws, size_t ws_size,
                              hipStream_t stream)
{
    const float* img = (const float*)d_in[0];
    const float* pts = (const float*)d_in[1];
    const float* nrm = (const float*)d_in[2];
    float* out      = (float*)d_out;
    float* partials = (float*)d_ws;

    const int N = in_sizes[1] / 2;                       // points is (N,2)
    const int W = (int)(sqrt((double)in_sizes[0]) + 0.5); // square image
    const int H = W;

    const int nBlocks = (N + BLOCK - 1) / BLOCK;
    const int padded  = (nBlocks + 63) & ~63;            // multiple of 64 for 16x4 WMMA tiles

    contour_main<<<nBlocks, BLOCK, 0, stream>>>(img, pts, nrm, partials,
                                                N, W, H, nBlocks, padded);
    contour_reduce<<<1, 32, 0, stream>>>(partials, padded, 1.f / (float)N, out);
}